// YOLOv3Loss_67018669687274
// MI455X (gfx1250) — compile-verified
//
#include <hip/hip_runtime.h>
#include <math.h>

#define IMGSZ 608.0f
#define NB 16
#define NT 50
#define NA 3
#define NC 80

typedef float v2f __attribute__((ext_vector_type(2)));
typedef float v8f __attribute__((ext_vector_type(8)));

__constant__ float g_anchors[3][3][2] = {
    {{116.f, 90.f}, {156.f, 198.f}, {373.f, 326.f}},
    {{30.f, 61.f}, {62.f, 45.f}, {59.f, 119.f}},
    {{10.f, 13.f}, {16.f, 30.f}, {33.f, 23.f}}};

// min(softplus(x), 100) == -max(log(sigmoid(-x)), -100) with stable softplus
__device__ inline float spc(float x) {
  float sp = fmaxf(x, 0.0f) + log1pf(expf(-fabsf(x)));
  return fminf(sp, 100.0f);
}

// Full-wave (32 lane) sum via V_WMMA_F32_16X16X4_F32.
// A (16x4 f32): v0 = p (K=0 lanes0-15 / K=2 lanes16-31), v1 = 0.
// B = all ones (4x16). D[m][n] = p[m] + p[m+16] for every n.
// Lane n<16 holds D[0..7][n] in its 8 D VGPRs, lane n>=16 holds D[8..15][n-16],
// so sum(D vgprs)@lane0 + sum(D vgprs)@lane16 = total.
// Must be called with EXEC all-ones (no divergence).
__device__ inline float wave_sum_wmma(float p) {
#if __has_builtin(__builtin_amdgcn_wmma_f32_16x16x4_f32)
  v2f a;
  a.x = p;
  a.y = 0.0f;
  v2f b;
  b.x = 1.0f;
  b.y = 1.0f;
  v8f c = {};
  v8f d = __builtin_amdgcn_wmma_f32_16x16x4_f32(false, a, false, b, (short)0, c,
                                                false, false);
  float s = d[0] + d[1] + d[2] + d[3] + d[4] + d[5] + d[6] + d[7];
  return __shfl(s, 0, 32) + __shfl(s, 16, 32);
#else
  for (int off = 16; off > 0; off >>= 1) p += __shfl_xor(p, off, 32);
  return p;
#endif
}

// sums layout (per scale s, stride 8): 0:Sx 1:Sy 2:Sw 3:Sh 4:Sconf_pos 5:Scls 6:Snoobj 7:n_pos
__global__ void zero_kernel(float* sums) {
  if (threadIdx.x < 24) sums[threadIdx.x] = 0.0f;
}

// One thread per (scale, batch). Serial over T for last-write-wins duplicate
// resolution (JAX duplicate scatter order is unspecified).
__global__ void assign_kernel(const float* __restrict__ labels,
                              int* __restrict__ packed, int* __restrict__ pose,
                              float4* __restrict__ tgt,
                              float* __restrict__ sums) {
  int tid = threadIdx.x;
  if (tid >= 3 * NB) return;
  int s = tid / NB;
  int b = tid - s * NB;
  const int Harr[3] = {19, 38, 76};
  int H = Harr[s], W = H;
  float aw[NA], ah[NA];
  for (int a = 0; a < NA; a++) {
    aw[a] = g_anchors[s][a][0] * (float)W / IMGSZ;
    ah[a] = g_anchors[s][a][1] * (float)H / IMGSZ;
  }
  // nlabel = count of rows whose raw 5-entry sum > 0; valid = (t < nlabel)
  int nl = 0;
  for (int t = 0; t < NT; t++) {
    const float* L = labels + (size_t)(b * NT + t) * 5;
    float sm = L[0] + L[1] + L[2] + L[3] + L[4];
    if (sm > 0.0f) nl++;
  }
  int base = (s * NB + b) * NT;
  float sc = (float)W / IMGSZ;
  for (int t = 0; t < NT; t++) {
    int pk = 0, po = 0;
    float4 tg = make_float4(0.f, 0.f, 0.f, 0.f);
    if (t < nl) {
      const float* L = labels + (size_t)(b * NT + t) * 5;
      int cls = ((int)L[0]) & 127;
      float gx = L[1] * sc, gy = L[2] * sc, gw = L[3] * sc, gh = L[4] * sc;
      int gi = (int)floorf(gx), gj = (int)floorf(gy);
      float denb = (gw + 1.f) * (gh + 1.f);
      int best = 0, ign = 0;
      float bi = -1.0f;
      for (int a = 0; a < NA; a++) {
        float inter = (fminf(gw, aw[a]) + 1.f) * (fminf(gh, ah[a]) + 1.f);
        float iou =
            inter / (denb + (aw[a] + 1.f) * (ah[a] + 1.f) - inter + 1e-16f);
        if (iou > bi) {  // strict > : first max wins ties (jnp.argmax)
          bi = iou;
          best = a;
        }
        if (iou > 0.5f) ign |= (1 << a);
      }
      bool inb = (gi >= 0 && gi < W && gj >= 0 && gj < H);  // mode='drop'
      if (inb) {
        pk = gi | (gj << 7) | (ign << 14) | (1 << 17);
        po = gi | (gj << 7) | (best << 14) | (cls << 16) | (1 << 23);
        tg = make_float4(gx - (float)gi, gy - (float)gj,
                         logf(gw / aw[best] + 1e-16f),
                         logf(gh / ah[best] + 1e-16f));
      }
    }
    packed[base + t] = pk;
    pose[base + t] = po;
    tgt[base + t] = tg;
  }
  // owner pass: last valid t with a given (best,gj,gi) cell wins; n_pos = #owners
  int npos = 0;
  for (int t = 0; t < NT; t++) {
    int e = pose[base + t];
    if ((e >> 23) & 1) {
      bool owner = true;
      for (int t2 = t + 1; t2 < NT; t2++) {
        int e2 = pose[base + t2];
        if (((e2 >> 23) & 1) && ((e2 & 0xFFFF) == (e & 0xFFFF))) {
          owner = false;
          break;
        }
      }
      if (owner) {
        pose[base + t] = e | (1 << 24);
        npos++;
      }
    }
  }
  atomicAdd(&sums[s * 8 + 7], (float)npos);
}

// One 128-thread block per (s,b,t). Threads 0..79: class BCEs (tcls = union of
// classes scattered into this cell); threads 80..84: x,y,w,h,conf positive terms.
__global__ void positive_loss_kernel(const float* __restrict__ p0,
                                     const float* __restrict__ p1,
                                     const float* __restrict__ p2,
                                     const int* __restrict__ pose,
                                     const float4* __restrict__ tgt,
                                     float* __restrict__ sums) {
  int t = blockIdx.x, b = blockIdx.y, s = blockIdx.z;
  const int Harr[3] = {19, 38, 76};
  int H = Harr[s], W = H;
  const float* pred = (s == 0) ? p0 : (s == 1) ? p1 : p2;
  int base = (s * NB + b) * NT;
  __shared__ int spose[NT];
  __shared__ float acc;
  int tid = threadIdx.x;
  if (tid < NT) spose[tid] = pose[base + tid];
  if (tid == 0) acc = 0.0f;
  __syncthreads();
  int e = spose[t];
  if (!((e >> 24) & 1)) return;  // uniform across block: not an owner cell
  int gi = e & 127, gj = (e >> 7) & 127, best = (e >> 14) & 3;
  size_t chstride = (size_t)H * W;
  size_t cellbase = ((size_t)(b * 255 + best * 85) * H + gj) * W + gi;
  if (tid < NC) {
    int mycell = e & 0xFFFF;
    int tc = 0;
    for (int t2 = 0; t2 < NT; t2++) {
      int e2 = spose[t2];
      if (((e2 >> 23) & 1) && ((e2 & 0xFFFF) == mycell) &&
          (((e2 >> 16) & 127) == tid))
        tc = 1;
    }
    float x = pred[cellbase + (size_t)(5 + tid) * chstride];
    float el = tc ? spc(-x) : spc(x);
    atomicAdd(&acc, el);
  } else if (tid < NC + 5) {
    int k = tid - NC;
    float4 tg = tgt[base + t];
    float x = pred[cellbase + (size_t)k * chstride];
    float el;
    if (k == 0)
      el = tg.x * spc(-x) + (1.0f - tg.x) * spc(x);  // BCE(sigmoid(x), tx)
    else if (k == 1)
      el = tg.y * spc(-x) + (1.0f - tg.y) * spc(x);
    else if (k == 2) {
      float d = x - tg.z;
      el = d * d;
    } else if (k == 3) {
      float d = x - tg.w;
      el = d * d;
    } else
      el = spc(-x);  // -log(sigmoid(conf)) clamped
    atomicAdd(&sums[s * 8 + k], el);
  }
  __syncthreads();
  if (tid == 0) atomicAdd(&sums[s * 8 + 5], acc);
}

// Dense pass: only the conf channel. noobj test against <=50 packed targets in
// LDS, then softplus; wave-level reduction via WMMA (EXEC all-ones, no early
// returns before the reduce).
__global__ void noobj_conf_kernel(const float* __restrict__ pred,
                                  const int* __restrict__ packed_s,
                                  float* __restrict__ sum_noobj, int H) {
  const int W = H;
  const int b = blockIdx.y;
  __shared__ int tinfo[NT];
  __shared__ float wsum[8];
  int tid = threadIdx.x;
  if (tid < NT) tinfo[tid] = packed_s[b * NT + tid];
  __syncthreads();
  int idx = blockIdx.x * blockDim.x + tid;
  int total = NA * H * W;
  float partial = 0.0f;
  if (idx < total) {
    int a = idx / (H * W);
    int r = idx - a * H * W;
    int j = r / W;
    int i = r - j * W;
    int noobj = 1;
    for (int t = 0; t < NT; t++) {
      int e = tinfo[t];
      if (((e >> 17) & 1) && ((e >> (14 + a)) & 1)) {
        if ((e & 127) == i && ((e >> 7) & 127) == j) noobj = 0;
      }
    }
    if (noobj) {
      float x = pred[((size_t)(b * 255 + a * 85 + 4) * H + j) * W + i];
      partial = spc(x);  // -clamp(log(1 - sigmoid(x)), -100)
    }
  }
  float ws = wave_sum_wmma(partial);
  int wid = tid >> 5;
  if ((tid & 31) == 0) wsum[wid] = ws;
  __syncthreads();
  if (tid == 0) {
    float sblk = 0.0f;
    for (int w = 0; w < 8; w++) sblk += wsum[w];
    atomicAdd(sum_noobj, sblk);
  }
}

__global__ void finalize_kernel(const float* __restrict__ sums,
                                float* __restrict__ out) {
  if (threadIdx.x != 0) return;
  const int Harr[3] = {19, 38, 76};
  float loss = 0.0f;
  for (int s = 0; s < 3; s++) {
    const float* S = sums + s * 8;
    float N = (float)(NB * NA * Harr[s] * Harr[s]);
    loss += (2.5f * (S[0] + S[1] + S[2] + S[3]) + S[4] + 0.5f * S[6]) / N +
            S[5] / (S[7] * (float)NC);
  }
  out[0] = loss;
}

extern "C" void kernel_launch(void* const* d_in, const int* in_sizes, int n_in,
                              void* d_out, int out_size, void* d_ws,
                              size_t ws_size, hipStream_t stream) {
  (void)in_sizes;
  (void)n_in;
  (void)out_size;
  (void)ws_size;
  const float* p0 = (const float*)d_in[0];
  const float* p1 = (const float*)d_in[1];
  const float* p2 = (const float*)d_in[2];
  const float* labels = (const float*)d_in[3];

  char* ws = (char*)d_ws;
  float* sums = (float*)ws;                              // 24 floats
  int* packed = (int*)(ws + 128);                        // 3*16*50 ints
  int* pose = (int*)(ws + 128 + 2400 * 4);               // 3*16*50 ints
  float4* tgt = (float4*)(ws + 128 + 4800 * 4);          // 3*16*50 float4 (16B aligned)

  zero_kernel<<<1, 32, 0, stream>>>(sums);
  assign_kernel<<<1, 64, 0, stream>>>(labels, packed, pose, tgt, sums);
  positive_loss_kernel<<<dim3(NT, NB, 3), 128, 0, stream>>>(p0, p1, p2, pose,
                                                            tgt, sums);
  const int Hs[3] = {19, 38, 76};
  const float* ps[3] = {p0, p1, p2};
  for (int s = 0; s < 3; s++) {
    int H = Hs[s];
    int total = NA * H * H;
    dim3 grid((total + 255) / 256, NB);
    noobj_conf_kernel<<<grid, 256, 0, stream>>>(ps[s], packed + s * NB * NT,
                                                sums + s * 8 + 6, H);
  }
  finalize_kernel<<<1, 32, 0, stream>>>(sums, (float*)d_out);
}